// GCNConv_7894149890261
// MI455X (gfx1250) — compile-verified
//
#include <hip/hip_runtime.h>

typedef __attribute__((ext_vector_type(2))) float v2f;
typedef __attribute__((ext_vector_type(8))) float v8f;

#define D_DIM   128
#define N_NODES 100000
#define N_EDGES 3200000

// ---------------------------------------------------------------------------
// Kernel 1: H = X @ W using V_WMMA_F32_16X16X4_F32 (full fp32 precision).
// grid.x = N_NODES/16 blocks; block = 256 threads = 8 waves.
// Wave w computes the 16x16 output tile at rows [blockIdx*16, +16),
// cols [w*16, +16). K dimension (128) swept in steps of 4 -> 32 WMMAs/wave.
// ---------------------------------------------------------------------------
__global__ __launch_bounds__(256) void gemm_xw_wmma(
    const float* __restrict__ X, const float* __restrict__ W,
    float* __restrict__ H) {
  const int lane  = threadIdx.x & 31;
  const int wave  = threadIdx.x >> 5;
  const int half  = lane >> 4;   // 0: lanes 0-15, 1: lanes 16-31
  const int l16   = lane & 15;
  const int mBase = blockIdx.x * 16;
  const int nBase = wave * 16;

  // A-fragment source row for this lane (M = mBase + l16)
  const float* Arow = X + (size_t)(mBase + l16) * D_DIM;
  // B-fragment column for this lane (N = nBase + l16)
  const float* Bcol = W + nBase + l16;

  v8f acc = {};
#pragma unroll 8
  for (int k0 = 0; k0 < D_DIM; k0 += 4) {
    const int k = k0 + 2 * half;
    v2f a, b;
    // A 16x4 f32: VGPR0 = K=k0+2*half, VGPR1 = K=k0+2*half+1
    a.x = Arow[k];
    a.y = Arow[k + 1];
    // B 4x16 f32: same K split across lane halves, N striped across lanes
    b.x = Bcol[(size_t)k * D_DIM];
    b.y = Bcol[(size_t)(k + 1) * D_DIM];
    acc = __builtin_amdgcn_wmma_f32_16x16x4_f32(
        /*neg_a=*/false, a, /*neg_b=*/false, b,
        /*c_mod=*/(short)0, acc, /*reuse_a=*/false, /*reuse_b=*/false);
  }

  // D layout: VGPR r -> M = r + 8*half, N = l16
  float* Hout = H + (size_t)mBase * D_DIM + nBase + l16;
#pragma unroll
  for (int r = 0; r < 8; ++r) {
    Hout[(size_t)(r + 8 * half) * D_DIM] = acc[r];
  }
}

// ---------------------------------------------------------------------------
// Kernel 2: zero the accumulator (d_out) with float4 stores.
// ---------------------------------------------------------------------------
__global__ void zero_out(float4* __restrict__ out, int n4) {
  int i = blockIdx.x * blockDim.x + threadIdx.x;
  if (i < n4) out[i] = make_float4(0.f, 0.f, 0.f, 0.f);
}

// ---------------------------------------------------------------------------
// Kernel 3: edge scatter. One wave per edge; lane l owns columns [4l, 4l+4).
// H row gather hits L2 (H = 51.2 MB < 192 MB L2). Non-returning fp32 atomics
// accumulate into out[row]. Edge arrays are streamed once -> non-temporal.
// ---------------------------------------------------------------------------
__global__ __launch_bounds__(256) void spmm_scatter(
    const int* __restrict__ erow, const int* __restrict__ ecol,
    const float* __restrict__ eval, const float* __restrict__ H,
    float* __restrict__ out) {
  const int lane = threadIdx.x & 31;
  const int e = blockIdx.x * 8 + (threadIdx.x >> 5);
  if (e >= N_EDGES) return;

  const int   r = __builtin_nontemporal_load(erow + e);
  const int   c = __builtin_nontemporal_load(ecol + e);
  const float v = __builtin_nontemporal_load(eval + e);

  const float4 h = ((const float4*)(H + (size_t)c * D_DIM))[lane];
  float* o = out + (size_t)r * D_DIM + lane * 4;
  atomicAdd(o + 0, v * h.x);
  atomicAdd(o + 1, v * h.y);
  atomicAdd(o + 2, v * h.z);
  atomicAdd(o + 3, v * h.w);
}

// ---------------------------------------------------------------------------
// Kernel 4: out = relu(out + b), vectorized float4 (row width 128 = 32 float4)
// ---------------------------------------------------------------------------
__global__ void bias_relu(float4* __restrict__ out,
                          const float4* __restrict__ b4, int n4) {
  int i = blockIdx.x * blockDim.x + threadIdx.x;
  if (i >= n4) return;
  float4 o = out[i];
  float4 b = b4[i & 31];  // 128 floats of bias = 32 float4, row-periodic
  o.x = fmaxf(o.x + b.x, 0.f);
  o.y = fmaxf(o.y + b.y, 0.f);
  o.z = fmaxf(o.z + b.z, 0.f);
  o.w = fmaxf(o.w + b.w, 0.f);
  out[i] = o;
}

// ---------------------------------------------------------------------------
// Inputs (setup_inputs order): edge_row, edge_col, edge_val, X, W, b
// ---------------------------------------------------------------------------
extern "C" void kernel_launch(void* const* d_in, const int* in_sizes, int n_in,
                              void* d_out, int out_size, void* d_ws, size_t ws_size,
                              hipStream_t stream) {
  const int*   erow = (const int*)d_in[0];
  const int*   ecol = (const int*)d_in[1];
  const float* eval = (const float*)d_in[2];
  const float* X    = (const float*)d_in[3];
  const float* W    = (const float*)d_in[4];
  const float* b    = (const float*)d_in[5];
  float* out = (float*)d_out;
  float* H   = (float*)d_ws;  // 100000*128*4 = 51.2 MB scratch

  // 1) H = X @ W  (fp32 WMMA)
  gemm_xw_wmma<<<N_NODES / 16, 256, 0, stream>>>(X, W, H);

  // 2) out = 0
  const int n4 = N_NODES * D_DIM / 4;  // 3.2M float4
  zero_out<<<(n4 + 255) / 256, 256, 0, stream>>>((float4*)out, n4);

  // 3) scatter-add edges (one wave per edge)
  spmm_scatter<<<(N_EDGES + 7) / 8, 256, 0, stream>>>(erow, ecol, eval, H, out);

  // 4) out = relu(out + b)
  bias_relu<<<(n4 + 255) / 256, 256, 0, stream>>>((float4*)out,
                                                  (const float4*)b, n4);
}